// Fuse_SS2D_46686294507589
// MI455X (gfx1250) — compile-verified
//
#include <hip/hip_runtime.h>
#include <hip/hip_bf16.h>

// SS2D fused: cross-scan -> WMMA fp32 projection -> selective scan -> cross-merge + LN
// Target: gfx1250 (wave32, WMMA)

#define B_  4
#define DI  192
#define LL  4096
#define NS  16
#define RR  6
#define KK  4
#define CP  38          // DT_RANK + 2*D_STATE
#define TL  64          // scan chunk length
#define TLP 65          // padded LDS stride (bank-conflict free)

typedef float v2f __attribute__((ext_vector_type(2)));
typedef float v8f __attribute__((ext_vector_type(8)));

// memory offset inside an HxW plane for scan-direction k at scan position l
__device__ __forceinline__ int scan_idx(int k, int l) {
    int ll = (k & 2) ? (LL - 1 - l) : l;
    if (k & 1) return ((ll & 63) << 6) | (ll >> 6);   // transpose: l = w*64+h -> mem h*64+w
    return ll;
}

// ---------------------------------------------------------------------------
// Kernel 1: x_dbl[b,k,c,l] = sum_d  xpw[k,c,d] * xs[b,k,d,l]   via fp32 WMMA
// One wave per 16x16 output tile. A = 16x4 of W, B = 4x16 of gathered x.
// ---------------------------------------------------------------------------
__global__ __launch_bounds__(256) void ss2d_proj_wmma(
    const float* __restrict__ x,
    const float* __restrict__ xpw,
    float* __restrict__ xdbl)
{
    int wave = (int)((blockIdx.x * blockDim.x + threadIdx.x) >> 5);
    int lane = threadIdx.x & 31;
    int nt = wave & 255;            // 256 N-tiles over L
    int mt = (wave >> 8) % 3;       // 3 M-tiles over CP=38 (padded to 48)
    int kd = (wave >> 8) / 3;       // b*4+k
    if (kd >= B_ * KK) return;
    int k = kd & 3;
    int b = kd >> 2;

    int m0 = mt << 4;
    int l0 = nt << 4;
    int hw = lane >> 4;             // half-wave selector (K split 0..1 / 2..3)
    int ln = lane & 15;

    const float* wbase = xpw + (size_t)k * CP * DI;
    const float* xb    = x + (size_t)b * DI * LL;

    int arow   = m0 + ln;
    int arow_c = arow < CP ? arow : (CP - 1);
    float amsk = arow < CP ? 1.0f : 0.0f;

    int midx = scan_idx(k, l0 + ln);  // gathered column index for this lane

    v8f acc = {0.f, 0.f, 0.f, 0.f, 0.f, 0.f, 0.f, 0.f};
    for (int k0 = 0; k0 < DI; k0 += 4) {
        // A fragment: lane holds W[arow][k0 + 2*hw + {0,1}]
        const float* ap = wbase + arow_c * DI + k0 + (hw << 1);
        v2f a; a.x = ap[0] * amsk; a.y = ap[1] * amsk;
        // B fragment: lane n holds X[k0 + 2*hw + {0,1}][l0+n]
        const float* bp = xb + (size_t)(k0 + (hw << 1)) * LL + midx;
        v2f bb; bb.x = bp[0]; bb.y = bp[LL];
        acc = __builtin_amdgcn_wmma_f32_16x16x4_f32(
            false, a, false, bb, (short)0, acc, false, false);
    }

    // C/D layout: reg i -> (m_local = i + 8*hw, col = ln)
    float* obase = xdbl + (size_t)kd * CP * LL + l0 + ln;
#pragma unroll
    for (int i = 0; i < 8; ++i) {
        int m = m0 + i + (hw << 3);
        if (m < CP) obase[(size_t)m * LL] = acc[i];
    }
}

// ---------------------------------------------------------------------------
// Kernel 2: selective scan. Block = 1024 threads = 64 channels x 16 states.
// Grid = B*K*3 (3 blocks of 64 channels each). dt-projection + softplus fused.
// ---------------------------------------------------------------------------
__global__ __launch_bounds__(1024) void ss2d_scan(
    const float* __restrict__ y,
    const float* __restrict__ dtw_g,    // (K, D, R)
    const float* __restrict__ dtb_g,    // (K, D)
    const float* __restrict__ Alogs,    // (K*D, N)
    const float* __restrict__ Ds_g,     // (K*D)
    const float* __restrict__ xdbl,     // (B*K, CP, L)
    float* __restrict__ outy)           // (B*K, D, L)
{
    __shared__ float sm_rows[CP * TLP];   // dts rows 0..5, B rows 6..21, C rows 22..37
    __shared__ float sm_u[64 * TLP];      // gathered y for 64 channels
    __shared__ float sm_y[64 * TLP];      // per-chunk outputs (coalesced store)

    int dblk = blockIdx.x % 3;
    int k    = (blockIdx.x / 3) & 3;
    int b    = blockIdx.x / 12;
    int kd   = b * 4 + k;

    int t  = threadIdx.x;
    int n  = t & 15;          // state index
    int dl = t >> 4;          // local channel 0..63
    int d  = dblk * 64 + dl;  // global channel
    int kdch = k * DI + d;

    float A_dn = -__expf(Alogs[(size_t)kdch * NS + n]);
    float dtw[RR];
#pragma unroll
    for (int r = 0; r < RR; ++r) dtw[r] = dtw_g[(size_t)kdch * RR + r];
    float dtb = dtb_g[kdch];
    float Dsd = Ds_g[kdch];

    const float* xdb = xdbl + (size_t)kd * CP * LL;
    const float* yb  = y + (size_t)b * DI * LL;

    float h = 0.0f;

    for (int c = 0; c < LL / TL; ++c) {
        int l0 = c * TL;
        // stage x_dbl rows (38 x 64)
        for (int i = t; i < CP * TL; i += 1024) {
            int r = i >> 6, col = i & 63;
            sm_rows[r * TLP + col] = xdb[(size_t)r * LL + l0 + col];
        }
        // stage gathered y (64 channels x 64)
        for (int i = t; i < 64 * TL; i += 1024) {
            int r = i >> 6, col = i & 63;
            sm_u[r * TLP + col] =
                yb[(size_t)(dblk * 64 + r) * LL + scan_idx(k, l0 + col)];
        }
        __syncthreads();

        for (int j = 0; j < TL; ++j) {
            // fused dt projection + softplus (uniform across the 16 state lanes)
            float s = dtb;
#pragma unroll
            for (int r = 0; r < RR; ++r) s += dtw[r] * sm_rows[r * TLP + j];
            float delta = (s > 20.0f) ? s : log1pf(__expf(s));

            float u  = sm_u[dl * TLP + j];
            float bv = sm_rows[(RR + n) * TLP + j];
            float cv = sm_rows[(RR + NS + n) * TLP + j];

            h = __expf(delta * A_dn) * h + (delta * u) * bv;
            float yv = h * cv;
            // reduce over 16 states within half-wave
            yv += __shfl_xor(yv, 1, 16);
            yv += __shfl_xor(yv, 2, 16);
            yv += __shfl_xor(yv, 4, 16);
            yv += __shfl_xor(yv, 8, 16);
            if (n == 0) sm_y[dl * TLP + j] = yv + Dsd * u;
        }
        __syncthreads();

        // coalesced chunk store
        for (int i = t; i < 64 * TL; i += 1024) {
            int r = i >> 6, col = i & 63;
            outy[((size_t)kd * DI + dblk * 64 + r) * LL + l0 + col] =
                sm_y[r * TLP + col];
        }
        // next iteration's staging writes sm_rows/sm_u (disjoint from sm_y);
        // the post-staging barrier orders sm_y reuse.
    }
}

// ---------------------------------------------------------------------------
// Kernel 3: cross-merge (4 directions) + LayerNorm over D=192.
// One 192-thread block per (b, l). Deterministic fixed-order sum (no atomics).
// ---------------------------------------------------------------------------
__global__ __launch_bounds__(192) void ss2d_merge_ln(
    const float* __restrict__ outy,
    const float* __restrict__ gamma,
    const float* __restrict__ beta,
    float* __restrict__ out)
{
    __shared__ float red[16];
    __shared__ float stats[2];

    int b  = blockIdx.x >> 12;
    int lm = blockIdx.x & 4095;           // l = h*64+w (row-major)
    int hh = lm >> 6, ww = lm & 63;
    int lt = (ww << 6) | hh;              // transposed position
    int d  = threadIdx.x;

    const size_t ks = (size_t)DI * LL;
    const float* base = outy + (size_t)b * KK * ks + (size_t)d * LL;

    float v = base[lm]                          // k=0
            + base[2 * ks + (LL - 1 - lm)]      // flip(k=2)
            + base[ks + lt]                     // transpose-back(k=1)
            + base[3 * ks + (LL - 1 - lt)];     // transpose-back(flip(k=3))

    float s = v, q = v * v;
#pragma unroll
    for (int o = 16; o; o >>= 1) {
        s += __shfl_xor(s, o, 32);
        q += __shfl_xor(q, o, 32);
    }
    int wv = d >> 5;
    if ((d & 31) == 0) { red[wv] = s; red[8 + wv] = q; }
    __syncthreads();
    if (d == 0) {
        float ts = 0.f, tq = 0.f;
        for (int i = 0; i < 6; ++i) { ts += red[i]; tq += red[8 + i]; }
        float mu  = ts / 192.0f;
        float var = tq / 192.0f - mu * mu;
        stats[0] = mu;
        stats[1] = rsqrtf(var + 1e-5f);
    }
    __syncthreads();
    float mu = stats[0], rstd = stats[1];
    out[(size_t)blockIdx.x * DI + d] = (v - mu) * rstd * gamma[d] + beta[d];
}

// ---------------------------------------------------------------------------
extern "C" void kernel_launch(void* const* d_in, const int* in_sizes, int n_in,
                              void* d_out, int out_size, void* d_ws, size_t ws_size,
                              hipStream_t stream) {
    const float* x   = (const float*)d_in[0];
    const float* y   = (const float*)d_in[1];
    const float* xpw = (const float*)d_in[2];
    const float* dtw = (const float*)d_in[3];
    const float* dtb = (const float*)d_in[4];
    const float* Al  = (const float*)d_in[5];
    const float* Ds  = (const float*)d_in[6];
    const float* gm  = (const float*)d_in[7];
    const float* bt  = (const float*)d_in[8];

    float* xdbl = (float*)d_ws;                                // (B*K, 38, L)  ~10 MB
    float* outy = xdbl + (size_t)B_ * KK * CP * LL;            // (B*K, 192, L) ~50 MB

    // 12288 output tiles, 8 waves (256 thr) per block -> 1536 blocks
    ss2d_proj_wmma<<<1536, 256, 0, stream>>>(x, xpw, xdbl);
    ss2d_scan<<<B_ * KK * 3, 1024, 0, stream>>>(y, dtw, dtb, Al, Ds, xdbl, outy);
    ss2d_merge_ln<<<B_ * LL, 192, 0, stream>>>(outy, gm, bt, (float*)d_out);
}